// GCN_79774722556359
// MI455X (gfx1250) — compile-verified
//
#include <hip/hip_runtime.h>

#define NUM_USER 50000
#define NUM_ITEM 50000
#define N_NODES  100000
#define DIM_E    64
#define DIM_FEAT 128
#define DIM_H    256          // 4 * DIM_E
#define N_EDGES  1600000
#define N_TILES  (NUM_ITEM / 16)   // 3125 row tiles of 16

typedef __attribute__((ext_vector_type(2))) float v2f;
typedef __attribute__((ext_vector_type(8))) float v8f;

// ---------------------------------------------------------------------------
// generic zero-fill
// ---------------------------------------------------------------------------
__global__ void k_zero(float* __restrict__ p, int n) {
    int i = blockIdx.x * blockDim.x + threadIdx.x;
    if (i < n) p[i] = 0.0f;
}

// ---------------------------------------------------------------------------
// MLP: temp = leaky_relu(F @ W1 + b1, 0.01) @ W2 + b2   via V_WMMA_F32_16X16X4_F32
// 1 wave == one 16-row tile of F. Hidden [16,256] staged in LDS per wave.
// Layouts per CDNA5 ISA 7.12.2 (32-bit A: lane M=l%16 holds K=2*(l/16)+{0,1};
// 32-bit B: lane N=l%16 holds same K pair; C/D: vgpr r -> M = r + 8*(l/16)).
// ---------------------------------------------------------------------------
__global__ __launch_bounds__(128) void k_mlp(
        const float* __restrict__ F,
        const float* __restrict__ W1, const float* __restrict__ b1,
        const float* __restrict__ W2, const float* __restrict__ b2,
        float* __restrict__ temp) {
    __shared__ float hid[4][16][DIM_H];          // 64 KB

    const int wave = threadIdx.x >> 5;
    const int lane = threadIdx.x & 31;
    const int lm   = lane & 15;                  // M (A) or N (B/C/D)
    const int lh   = lane >> 4;                  // K-pair selector / row-half
    const int tile = blockIdx.x * 4 + wave;      // whole-wave guard: EXEC stays all-1s
    if (tile >= N_TILES) return;
    const int r0 = tile * 16;

    // ---- layer 1: hidden[16,256] = leaky_relu(F_tile @ W1 + b1) ----
    for (int nt = 0; nt < DIM_H / 16; ++nt) {
        const int N = nt * 16 + lm;
        v8f acc = {};
        #pragma unroll 4
        for (int kb = 0; kb < DIM_FEAT / 4; ++kb) {
            const int k0 = kb * 4 + lh * 2;
            v2f a = *(const v2f*)(F + (r0 + lm) * DIM_FEAT + k0);
            v2f b;
            b.x = W1[(k0    ) * DIM_H + N];
            b.y = W1[(k0 + 1) * DIM_H + N];
            acc = __builtin_amdgcn_wmma_f32_16x16x4_f32(
                      false, a, false, b, (short)0, acc, false, false);
        }
        const float bias = b1[N];
        #pragma unroll
        for (int r = 0; r < 8; ++r) {
            float hv = acc[r] + bias;
            hv = (hv > 0.0f) ? hv : 0.01f * hv;   // leaky_relu(0.01)
            hid[wave][r + 8 * lh][N] = hv;
        }
    }
    // no barrier needed: each wave consumes only its own LDS slice
    // (hardware DScnt ordering covers intra-wave LDS RAW)

    // ---- layer 2: temp_tile = hidden @ W2 + b2 ----
    for (int nt = 0; nt < DIM_E / 16; ++nt) {
        const int N = nt * 16 + lm;
        v8f acc = {};
        #pragma unroll 4
        for (int kb = 0; kb < DIM_H / 4; ++kb) {
            const int k0 = kb * 4 + lh * 2;
            v2f a = *(const v2f*)(&hid[wave][lm][k0]);
            v2f b;
            b.x = W2[(k0    ) * DIM_E + N];
            b.y = W2[(k0 + 1) * DIM_E + N];
            acc = __builtin_amdgcn_wmma_f32_16x16x4_f32(
                      false, a, false, b, (short)0, acc, false, false);
        }
        const float bias = b2[N];
        #pragma unroll
        for (int r = 0; r < 8; ++r)
            temp[(r0 + r + 8 * lh) * DIM_E + N] = acc[r] + bias;
    }
}

// ---------------------------------------------------------------------------
// L2-normalize rows of concat(preference, temp) -> xn.  One wave per node.
// ---------------------------------------------------------------------------
__global__ void k_normalize(const float* __restrict__ pref,
                            const float* __restrict__ temp,
                            float* __restrict__ xn) {
    const int wave = threadIdx.x >> 5;
    const int lane = threadIdx.x & 31;
    const int node = blockIdx.x * (blockDim.x >> 5) + wave;
    if (node >= N_NODES) return;
    const float* src = (node < NUM_USER) ? (pref + (size_t)node * DIM_E)
                                         : (temp + (size_t)(node - NUM_USER) * DIM_E);
    float v0 = src[lane];
    float v1 = src[lane + 32];
    float s = v0 * v0 + v1 * v1;
    #pragma unroll
    for (int off = 16; off > 0; off >>= 1) s += __shfl_xor(s, off, 32);
    const float inv = 1.0f / fmaxf(sqrtf(s), 1e-12f);
    xn[(size_t)node * DIM_E + lane]      = v0 * inv;
    xn[(size_t)node * DIM_E + lane + 32] = v1 * inv;
}

// ---------------------------------------------------------------------------
// deg[row] += 1 for non-self-loop edges (matches segment_sum(emask, row))
// ---------------------------------------------------------------------------
__global__ void k_degree(const int* __restrict__ ei, float* __restrict__ deg) {
    int e = blockIdx.x * blockDim.x + threadIdx.x;
    if (e >= N_EDGES) return;
    int r = ei[e];
    int c = ei[N_EDGES + e];
    if (r != c) atomicAdd(deg + r, 1.0f);
}

// in-place: deg -> deg^-0.5 (0 where deg==0)
__global__ void k_dis(float* __restrict__ deg) {
    int n = blockIdx.x * blockDim.x + threadIdx.x;
    if (n >= N_NODES) return;
    float d = deg[n];
    deg[n] = (d > 0.0f) ? rsqrtf(d) : 0.0f;
}

// ---------------------------------------------------------------------------
// GCN conv: out[col] += x[row] * dis[row]*dis[col].  16 lanes/edge, float4
// gathers (global_load_b128) + global_atomic_add_f32 scatters; x and dis are
// L2-resident (25.6 MB << 192 MB), so this is L2-atomic bound by design.
// ---------------------------------------------------------------------------
__global__ void k_conv(const int* __restrict__ ei, const float* __restrict__ dis,
                       const float* __restrict__ x, float* __restrict__ out) {
    int tid = blockIdx.x * blockDim.x + threadIdx.x;
    int e = tid >> 4;
    int q = tid & 15;
    if (e >= N_EDGES) return;
    int r = ei[e];
    int c = ei[N_EDGES + e];
    if (r == c) return;
    float nrm = dis[r] * dis[c];
    if (nrm == 0.0f) return;
    const float4 v = *(const float4*)(x + (size_t)r * DIM_E + q * 4);
    float* o = out + (size_t)c * DIM_E + q * 4;
    atomicAdd(o + 0, v.x * nrm);
    atomicAdd(o + 1, v.y * nrm);
    atomicAdd(o + 2, v.z * nrm);
    atomicAdd(o + 3, v.w * nrm);
}

// ---------------------------------------------------------------------------
// x_hat = xn + h + h1; also copy preference to output tail. float4 streams.
// ---------------------------------------------------------------------------
__global__ void k_final(const float* __restrict__ xn, const float* __restrict__ h,
                        const float* __restrict__ h1, const float* __restrict__ pref,
                        float* __restrict__ out) {
    const int n4 = N_NODES * DIM_E / 4;      // 1.6M float4
    const int p4 = NUM_USER * DIM_E / 4;     // 0.8M float4
    int i = blockIdx.x * blockDim.x + threadIdx.x;
    if (i >= n4) return;
    float4 a = ((const float4*)xn)[i];
    float4 b = ((const float4*)h)[i];
    float4 c = ((const float4*)h1)[i];
    float4 rr;
    rr.x = a.x + b.x + c.x;
    rr.y = a.y + b.y + c.y;
    rr.z = a.z + b.z + c.z;
    rr.w = a.w + b.w + c.w;
    ((float4*)out)[i] = rr;
    if (i < p4) ((float4*)out)[n4 + i] = ((const float4*)pref)[i];
}

// ---------------------------------------------------------------------------
extern "C" void kernel_launch(void* const* d_in, const int* in_sizes, int n_in,
                              void* d_out, int out_size, void* d_ws, size_t ws_size,
                              hipStream_t stream) {
    const int*   ei   = (const int*)  d_in[0];   // [2, N_EDGES]
    const float* F    = (const float*)d_in[1];   // [50000, 128]
    const float* pref = (const float*)d_in[2];   // [50000, 64]
    const float* W1   = (const float*)d_in[3];   // [128, 256]
    const float* b1   = (const float*)d_in[4];   // [256]
    const float* W2   = (const float*)d_in[5];   // [256, 64]
    const float* b2   = (const float*)d_in[6];   // [64]
    float*       out  = (float*)d_out;

    // workspace layout (floats): xn | h | h1 | temp | deg   (~90.0 MB)
    float* ws   = (float*)d_ws;
    float* xn   = ws;
    float* h    = xn   + (size_t)N_NODES * DIM_E;
    float* h1   = h    + (size_t)N_NODES * DIM_E;
    float* temp = h1   + (size_t)N_NODES * DIM_E;
    float* deg  = temp + (size_t)NUM_ITEM * DIM_E;

    const int T = 256;
    // zero scatter targets + degree accumulator (h,h1 are contiguous)
    {
        int n = 2 * N_NODES * DIM_E;
        k_zero<<<(n + T - 1) / T, T, 0, stream>>>(h, n);
        k_zero<<<(N_NODES + T - 1) / T, T, 0, stream>>>(deg, N_NODES);
    }
    // MLP (WMMA): 4 tiles/block of 128 threads
    k_mlp<<<(N_TILES + 3) / 4, 128, 0, stream>>>(F, W1, b1, W2, b2, temp);
    // normalize concat(pref, temp)
    k_normalize<<<(N_NODES + 7) / 8, T, 0, stream>>>(pref, temp, xn);
    // degrees -> dis
    k_degree<<<(N_EDGES + T - 1) / T, T, 0, stream>>>(ei, deg);
    k_dis<<<(N_NODES + T - 1) / T, T, 0, stream>>>(deg);
    // two propagation hops
    {
        int nthr = N_EDGES * 16;
        k_conv<<<(nthr + T - 1) / T, T, 0, stream>>>(ei, deg, xn, h);
        k_conv<<<(nthr + T - 1) / T, T, 0, stream>>>(ei, deg, h, h1);
    }
    // epilogue
    k_final<<<(N_NODES * DIM_E / 4 + T - 1) / T, T, 0, stream>>>(xn, h, h1, pref, out);
}